// LGI_21062519619783
// MI455X (gfx1250) — compile-verified
//
#include <hip/hip_runtime.h>

// ---------------------------------------------------------------------------
// Problem constants (from reference): batch_x [B=16, C=3, T=600, H=72, W=72]
// ---------------------------------------------------------------------------
#define B_   16
#define C_   3
#define T_   600
#define HW_  5184          // 72*72, contiguous innermost per (b,c,t)
#define RP_  16            // padded rows per batch in the trace workspace

typedef float v2f __attribute__((ext_vector_type(2)));
typedef float v4f __attribute__((ext_vector_type(4)));
typedef float v8f __attribute__((ext_vector_type(8)));

// ---------------------------------------------------------------------------
// Stage 1: spatial mean over H*W for each (b,c,t).  Pure HBM streaming:
// 597 MB read at 23.3 TB/s -> ~26us floor; everything else is noise.
// float4 non-temporal loads (data = 3x L2 capacity, don't pollute),
// wave32 shuffle reduction + tiny LDS cross-wave combine.
// Output goes into a 16-row padded layout: xp[b][row][t], rows 3..15 are
// pre-zeroed by a memset so stage 2's WMMA operand loads are unconditional.
// ---------------------------------------------------------------------------
__global__ __launch_bounds__(256)
void spatial_mean_kernel(const float* __restrict__ in, float* __restrict__ xp) {
  const int idx = blockIdx.x;                       // flat (b,c,t), 0..28799
  const v4f* p = reinterpret_cast<const v4f*>(in + (size_t)idx * HW_);
  float s = 0.0f;
  // HW_/4 = 1296 float4 per block; ~5 per thread
  for (int i = threadIdx.x; i < HW_ / 4; i += 256) {
    v4f v = __builtin_nontemporal_load(&p[i]);
    s += (v.x + v.y) + (v.z + v.w);
  }
  // wave32 reduction
  #pragma unroll
  for (int off = 16; off > 0; off >>= 1) s += __shfl_xor(s, off, 32);
  __shared__ float red[8];
  const int lane = threadIdx.x & 31;
  const int wid  = threadIdx.x >> 5;
  if (lane == 0) red[wid] = s;
  __syncthreads();
  if (threadIdx.x == 0) {
    float t = 0.0f;
    #pragma unroll
    for (int w = 0; w < 8; ++w) t += red[w];
    const int b = idx / (C_ * T_);
    const int r = idx % (C_ * T_);                  // c*T_ + t
    xp[b * (RP_ * T_) + r] = t * (1.0f / (float)HW_);
  }
}

// ---------------------------------------------------------------------------
// Stage 2: per batch b, X = xp[b] is (16-padded)x600 with 3 live rows.
// Dominant left singular vector of X == dominant eigenvector of the 3x3 PSD
// Gram G = X X^T.  Build G with V_WMMA_F32_16X16X4_F32 over K chunks of 4:
//   A (16x4 f32):  lane l: M = l&15, holds K = 2*(l>>4), 2*(l>>4)+1
//   B (4x16 f32):  lane l: N = l&15, holds K = 2*(l>>4), 2*(l>>4)+1
// With B[k][n] = X[n][k] the per-lane values are IDENTICAL to A's, so one
// 8B load (v2f) feeds both operands; padded rows contribute zeros.
// D[m][n] = sum_k X[m,k] X[n,k] = G.  Each lane then extracts G via shuffles,
// power-iterates the 3x3 redundantly (sign irrelevant: P = I - s s^T) and
// projects:  out[b,n] = x1[n] - s1 * (s . x[:,n]).
// One wave per block: EXEC all ones (required for WMMA), zero divergence.
// ---------------------------------------------------------------------------
__global__ __launch_bounds__(32)
void gram_project_kernel(const float* __restrict__ xp, float* __restrict__ out) {
  const int b = blockIdx.x;
  const float* xb = xp + b * (RP_ * T_);
  const int lane  = threadIdx.x;        // 0..31
  const int row   = lane & 15;          // A: M-row / B: N-col
  const int khalf = lane >> 4;          // K pair {0,1} vs {2,3}
  const float* prow = xb + row * T_ + 2 * khalf;   // 8B-aligned (k even)

  v8f acc = {0.f, 0.f, 0.f, 0.f, 0.f, 0.f, 0.f, 0.f};
  #pragma unroll 5
  for (int k0 = 0; k0 < T_; k0 += 4) {
    const v2f a = *reinterpret_cast<const v2f*>(prow + k0);  // global_load_b64
    acc = __builtin_amdgcn_wmma_f32_16x16x4_f32(
        /*neg_a=*/false, a, /*neg_b=*/false, a,
        /*c_mod=*/(short)0, acc, /*reuse_a=*/false, /*reuse_b=*/false);
  }

  // Extract symmetric 3x3 Gram: G[m][n] lives in acc[m] at lane n (m<8, n<16)
  const float g00 = __shfl(acc[0], 0, 32);
  const float g01 = __shfl(acc[0], 1, 32);
  const float g02 = __shfl(acc[0], 2, 32);
  const float g11 = __shfl(acc[1], 1, 32);
  const float g12 = __shfl(acc[1], 2, 32);
  const float g22 = __shfl(acc[2], 2, 32);

  // Power iteration for dominant eigenvector (all lanes redundantly; G is PSD)
  float s0 = 1.0f, s1 = 0.5f, s2 = 0.25f;
  #pragma unroll 1
  for (int it = 0; it < 64; ++it) {
    const float t0 = g00 * s0 + g01 * s1 + g02 * s2;
    const float t1 = g01 * s0 + g11 * s1 + g12 * s2;
    const float t2 = g02 * s0 + g12 * s1 + g22 * s2;
    const float inv = rsqrtf(t0 * t0 + t1 * t1 + t2 * t2 + 1e-30f);
    s0 = t0 * inv; s1 = t1 * inv; s2 = t2 * inv;
  }

  // Project and take green channel: out[b,n] = x1 - s1 * (s . x[:,n])
  for (int n = lane; n < T_; n += 32) {
    const float x0 = xb[n];
    const float x1 = xb[T_ + n];
    const float x2 = xb[2 * T_ + n];
    const float dot = s0 * x0 + s1 * x1 + s2 * x2;
    out[b * T_ + n] = x1 - s1 * dot;
  }
}

// ---------------------------------------------------------------------------
// Launch: zero the padded trace buffer (rows 3..15 must be 0 for the WMMA
// operands), stage 1 fills rows 0..2 with the 16*3*600 means, stage 2
// consumes.  hipMemsetAsync on `stream` is graph-capture-safe.
// ---------------------------------------------------------------------------
extern "C" void kernel_launch(void* const* d_in, const int* in_sizes, int n_in,
                              void* d_out, int out_size, void* d_ws, size_t ws_size,
                              hipStream_t stream) {
  (void)in_sizes; (void)n_in; (void)out_size; (void)ws_size;
  const float* in = (const float*)d_in[0];
  float* out = (float*)d_out;
  float* xp  = (float*)d_ws;                 // B_*RP_*T_ = 153,600 floats (600 KB)

  hipMemsetAsync(xp, 0, (size_t)B_ * RP_ * T_ * sizeof(float), stream);
  spatial_mean_kernel<<<B_ * C_ * T_, 256, 0, stream>>>(in, xp);
  gram_project_kernel<<<B_, 32, 0, stream>>>(xp, out);
}